// ReformerLayer_45423574122705
// MI455X (gfx1250) — compile-verified
//
#include <hip/hip_runtime.h>
#include <hip/hip_bf16.h>

// ---------------- problem constants ----------------
#define BATCH 4
#define NTOK 4096
#define DM 512
#define NH 8
#define DH 64
#define BHD (BATCH*NH)            // 32
#define NHASH 4
#define NROT 32                   // n_buckets/2
#define BS 64                     // bucket (chunk) size
#define CPR (NTOK/BS)             // chunks per round = 64
#define NCHUNK (NHASH*CPR)        // 256 chunks per bh
#define SELF_ATTN_VALUE (-5.0e4f)

typedef __attribute__((ext_vector_type(2))) float v2f;
typedef __attribute__((ext_vector_type(8))) float v8f;

__device__ __forceinline__ v8f wmma_f32x4(v2f a, v2f b, v8f c) {
  // D(16x16,f32) = A(16x4,f32) x B(4x16,f32) + C
  return __builtin_amdgcn_wmma_f32_16x16x4_f32(false, a, false, b, (short)0, c,
                                               false, false);
}

// ---------------- CDNA5 async global->LDS copy (ASYNCcnt-tracked) ----------------
#if defined(__HIP_DEVICE_COMPILE__) &&                                         \
    __has_builtin(__builtin_amdgcn_global_load_async_to_lds_b128) &&           \
    __has_builtin(__builtin_amdgcn_s_wait_asynccnt)
#define HAVE_ASYNC_LDS 1
#else
#define HAVE_ASYNC_LDS 0
#endif

#if HAVE_ASYNC_LDS
// Builtin signature is (int4 AS1*, int4 AS3*, imm offset, imm cpol) with
// GCC-style vector pointee (per hipcc diagnostic).
typedef int v4i_gcc __attribute__((vector_size(16)));
typedef __attribute__((address_space(1))) v4i_gcc* as1_v4i;
typedef __attribute__((address_space(3))) v4i_gcc* as3_v4i;
// lds: generic pointer to __shared__; low 32 bits of the flat shared-aperture
// address are the wave-relative LDS byte offset (ISA 10.2), so truncation gives
// a valid AS3 address. Global generic pointers bit-match AS1.
__device__ __forceinline__ void async_ld16(void* lds, const void* g) {
  __builtin_amdgcn_global_load_async_to_lds_b128(
      (as1_v4i)(unsigned long long)g,
      (as3_v4i)(unsigned)(unsigned long long)lds, 0, 0);
}
__device__ __forceinline__ void async_wait0() {
  __builtin_amdgcn_s_wait_asynccnt(0);
}
#endif

// ---------------- utility ----------------
__global__ void zero_f32(float* __restrict__ p, int n) {
  int i = blockIdx.x * blockDim.x + threadIdx.x;
  if (i < n) p[i] = 0.0f;
}

// ---------------- stage 1: qk/v projections ----------------
// X:(16384,512)  W:(512,512)  out in merged layout (BH, NTOK, DH)
__global__ void gemm_qkv(const float* __restrict__ X,
                         const float* __restrict__ Wqk,
                         const float* __restrict__ Wv,
                         float* __restrict__ qk_m,
                         float* __restrict__ v_m) {
  const int lane = threadIdx.x & 31;
  const int wave = threadIdx.x >> 5;
  const int half = lane >> 4;
  const int l16  = lane & 15;
  const int mBase = blockIdx.x * 16;
  const int nBase = blockIdx.y * 128 + wave * 16;
  const float* __restrict__ W = blockIdx.z ? Wv : Wqk;
  float* __restrict__ Out = blockIdx.z ? v_m : qk_m;

  const int mA = mBase + l16;          // A operand row for this lane
  const int n  = nBase + l16;          // B/D column for this lane
  const float* __restrict__ arow = X + (size_t)mA * DM;

  v8f acc = {};
  for (int k = 0; k < DM; k += 4) {
    const int ka = k + 2 * half;
    v2f a, b;
    a.x = arow[ka];
    a.y = arow[ka + 1];
    b.x = W[(size_t)ka * DM + n];
    b.y = W[(size_t)(ka + 1) * DM + n];
    acc = wmma_f32x4(a, b, acc);
  }
  const int head = n >> 6, e = n & 63;
#pragma unroll
  for (int r = 0; r < 8; ++r) {
    const int m = mBase + r + 8 * half;
    const int bidx = m / NTOK, t = m % NTOK;
    Out[(((size_t)bidx * NH + head) * NTOK + t) * DH + e] = acc[r];
  }
}

// ---------------- stage 2: LSH hashing ----------------
// rot:(DH, NHASH, NROT)  buckets:(BH, NHASH, NTOK), value in [0,64)
__global__ void hash_kernel(const float* __restrict__ qk_m,
                            const float* __restrict__ rot,
                            unsigned* __restrict__ buckets) {
  __shared__ float rotS[DH * NHASH * NROT];  // 32KB
#if HAVE_ASYNC_LDS
  {
    const char* gsrc = (const char*)rot;
    char* ldst = (char*)rotS;
    for (int i = threadIdx.x; i < (DH * NHASH * NROT) / 4; i += blockDim.x)
      async_ld16(ldst + 16 * i, gsrc + 16 * i);
    async_wait0();
  }
#else
  for (int i = threadIdx.x; i < DH * NHASH * NROT; i += blockDim.x)
    rotS[i] = rot[i];
#endif
  __syncthreads();

  const int gid = blockIdx.x * blockDim.x + threadIdx.x;  // BH*NTOK threads
  const int bh = gid / NTOK, t = gid % NTOK;
  const float4* __restrict__ src =
      (const float4*)(qk_m + ((size_t)bh * NTOK + t) * DH);
  float q[DH];
#pragma unroll
  for (int e4 = 0; e4 < DH / 4; ++e4) {
    float4 x = src[e4];
    q[4 * e4 + 0] = x.x; q[4 * e4 + 1] = x.y;
    q[4 * e4 + 2] = x.z; q[4 * e4 + 3] = x.w;
  }
  for (int h = 0; h < NHASH; ++h) {
    float bv1 = -3.4e38f, bv2 = -3.4e38f;
    int bi1 = 0, bi2 = 0;
    for (int i = 0; i < NROT; ++i) {
      float s = 0.0f;
#pragma unroll
      for (int e = 0; e < DH; ++e) s += q[e] * rotS[(e * NHASH + h) * NROT + i];
      if (s > bv1) { bv1 = s; bi1 = i; }
      if (-s > bv2) { bv2 = -s; bi2 = i; }
    }
    // first-occurrence argmax over [rot, -rot]
    const unsigned b = (bv1 >= bv2) ? (unsigned)bi1 : (unsigned)(NROT + bi2);
    buckets[((size_t)bh * NHASH + h) * NTOK + t] = b;
  }
}

// ---------------- stage 3: stable counting sort per (bh, round) ----------------
// st[bh][r][p] = original position t (0..4095) of p-th sorted element.
__global__ void sort_kernel(const unsigned* __restrict__ buckets,
                            unsigned* __restrict__ st) {
  __shared__ unsigned hist[64][64];  // [thread][bucket]
  __shared__ unsigned base[64];
  const int bhr = blockIdx.x;  // bh*NHASH + r
  const unsigned* __restrict__ bk = buckets + (size_t)bhr * NTOK;
  unsigned* __restrict__ out = st + (size_t)bhr * NTOK;
  const int tau = threadIdx.x;  // 64 threads

  for (int b = 0; b < 64; ++b) hist[tau][b] = 0;
  __syncthreads();
  for (int i = 0; i < 64; ++i) hist[tau][bk[tau * 64 + i]]++;
  __syncthreads();
  {  // per-bucket totals
    unsigned s = 0;
    for (int tt = 0; tt < 64; ++tt) s += hist[tt][tau];
    base[tau] = s;
  }
  __syncthreads();
  if (tau == 0) {  // exclusive scan over buckets
    unsigned run = 0;
    for (int b = 0; b < 64; ++b) { unsigned c = base[b]; base[b] = run; run += c; }
  }
  __syncthreads();
  {  // exclusive scan over threads within bucket, folded with base
    unsigned run = base[tau];
    for (int tt = 0; tt < 64; ++tt) { unsigned c = hist[tt][tau]; hist[tt][tau] = run; run += c; }
  }
  __syncthreads();
  for (int i = 0; i < 64; ++i) {
    const unsigned b = bk[tau * 64 + i];
    out[hist[tau][b]++] = (unsigned)(tau * 64 + i);
  }
}

// ---------------- stage 4/6: chunked LSH attention ----------------
// PB=false: compute per-row logsumexp -> logits.
// PB=true : recompute probs, bo = probs @ v, ctx += weight * bo (one round per launch).
template <bool PB>
__global__ void chunk_kernel(const float* __restrict__ qk_m,
                             const float* __restrict__ v_m,
                             const unsigned* __restrict__ st,
                             const float* __restrict__ logits_in,
                             float* __restrict__ logits_out,
                             const float* __restrict__ weight,
                             float* __restrict__ ctx,
                             int chunksPerBH, int cBase) {
  extern __shared__ float smem[];
  float* kT = smem;                              // [DH][128] -> reused as dots[64][128]
  float* qcur = smem + 8192;                     // [64][DH]
  float* vbuf = smem + 12288;                    // [128][DH]  (PB only)
  unsigned* stkv = (unsigned*)(smem + (PB ? 20480 : 12288));  // [128]
  float* lseS = (float*)(stkv + 128);            // [64]  (PB)
  float* wS = lseS + 64;                         // [64]  (PB)

  const int tid = threadIdx.x;
  const int bh = blockIdx.x / chunksPerBH;
  const int c = cBase + blockIdx.x % chunksPerBH;
  const int r = c / CPR;
  const int cp = (c + NCHUNK - 1) % NCHUNK;      // look-one-back chunk
  const int rp = cp / CPR;

  if (tid < BS) {
    stkv[tid]      = st[((size_t)(bh * NHASH + r)  * NTOK) + (c  % CPR) * BS + tid];
    stkv[BS + tid] = st[((size_t)(bh * NHASH + rp) * NTOK) + (cp % CPR) * BS + tid];
  }
  __syncthreads();

  // gather: threads 0..127 -> keys (normalized, transposed); 0..63 also raw q
  if (tid < 128) {
    const int j = tid;
    const float4* __restrict__ src =
        (const float4*)(qk_m + ((size_t)bh * NTOK + stkv[j]) * DH);
    float rowv[DH];
    float ss = 0.0f;
#pragma unroll
    for (int e4 = 0; e4 < DH / 4; ++e4) {
      float4 x = src[e4];
      rowv[4 * e4 + 0] = x.x; rowv[4 * e4 + 1] = x.y;
      rowv[4 * e4 + 2] = x.z; rowv[4 * e4 + 3] = x.w;
      ss += x.x * x.x + x.y * x.y + x.z * x.z + x.w * x.w;
    }
    const float inv = 1.0f / fmaxf(sqrtf(ss), 1e-12f);
#pragma unroll
    for (int e = 0; e < DH; ++e) kT[e * 128 + j] = rowv[e] * inv;
    if (j < BS) {
#pragma unroll
      for (int e = 0; e < DH; ++e) qcur[j * DH + e] = rowv[e];
    }
  } else if (PB) {
    const int j = tid - 128;  // v rows 0..127, gathered straight into LDS
    const char* gsrc = (const char*)(v_m + ((size_t)bh * NTOK + stkv[j]) * DH);
    char* ldst = (char*)(vbuf + j * DH);
#if HAVE_ASYNC_LDS
#pragma unroll
    for (int e4 = 0; e4 < DH / 4; ++e4)
      async_ld16(ldst + 16 * e4, gsrc + 16 * e4);
#else
    const float4* __restrict__ src4 = (const float4*)gsrc;
    float4* dst4 = (float4*)ldst;
#pragma unroll
    for (int e4 = 0; e4 < DH / 4; ++e4) dst4[e4] = src4[e4];
#endif
  }
#if HAVE_ASYNC_LDS
  if (PB) async_wait0();  // issuing waves drain their ASYNCcnt before the barrier
#endif
  __syncthreads();

  // dots(64x128) = qcur(64x64) @ kT(64x128), f32 WMMA 16x16x4 tiles
  const int wave = tid >> 5, lane = tid & 31, half = lane >> 4, l16 = lane & 15;
  const int mT = wave >> 1;                 // 0..3
  const int nOff = (wave & 1) * 64;         // wave owns 4 N-tiles
  const int mRow = mT * 16 + l16;
  v8f d0 = {}, d1 = {}, d2 = {}, d3 = {};
  for (int k = 0; k < DH; k += 4) {
    const int kk = k + 2 * half;
    v2f a; a.x = qcur[mRow * DH + kk]; a.y = qcur[mRow * DH + kk + 1];
    const int nc = nOff + l16;
    v2f b;
    b.x = kT[kk * 128 + nc +  0]; b.y = kT[(kk + 1) * 128 + nc +  0];
    d0 = wmma_f32x4(a, b, d0);
    b.x = kT[kk * 128 + nc + 16]; b.y = kT[(kk + 1) * 128 + nc + 16];
    d1 = wmma_f32x4(a, b, d1);
    b.x = kT[kk * 128 + nc + 32]; b.y = kT[(kk + 1) * 128 + nc + 32];
    d2 = wmma_f32x4(a, b, d2);
    b.x = kT[kk * 128 + nc + 48]; b.y = kT[(kk + 1) * 128 + nc + 48];
    d3 = wmma_f32x4(a, b, d3);
  }
  __syncthreads();  // everyone done reading kT; reuse it as dots

  float* dots = kT;
  const float scale = 0.125f;  // 64^-0.5
#pragma unroll
  for (int rr = 0; rr < 8; ++rr) {
    const int m = mT * 16 + rr + 8 * half;
    const unsigned tq = stkv[m];
    const int n0 = nOff + l16;
    float x0 = d0[rr] * scale; if (tq == stkv[n0 +  0]) x0 = SELF_ATTN_VALUE;
    float x1 = d1[rr] * scale; if (tq == stkv[n0 + 16]) x1 = SELF_ATTN_VALUE;
    float x2 = d2[rr] * scale; if (tq == stkv[n0 + 32]) x2 = SELF_ATTN_VALUE;
    float x3 = d3[rr] * scale; if (tq == stkv[n0 + 48]) x3 = SELF_ATTN_VALUE;
    dots[m * 128 + n0 +  0] = x0;
    dots[m * 128 + n0 + 16] = x1;
    dots[m * 128 + n0 + 32] = x2;
    dots[m * 128 + n0 + 48] = x3;
  }
  __syncthreads();

  if (!PB) {
    if (tid < BS) {
      const int m = tid;
      float mx = -3.4e38f;
      for (int j = 0; j < 128; ++j) mx = fmaxf(mx, dots[m * 128 + j]);
      float s = 0.0f;
      for (int j = 0; j < 128; ++j) s += expf(dots[m * 128 + j] - mx);
      logits_out[((size_t)(bh * NHASH + r) * NTOK) + stkv[m]] = mx + logf(s);
    }
    return;
  }

  // phase B: probs = exp(dots - lse), bo = probs @ v, ctx += w * bo
  if (tid < BS) {
    const size_t gi = ((size_t)(bh * NHASH + r) * NTOK) + stkv[tid];
    lseS[tid] = logits_in[gi];
    wS[tid] = weight[gi];
  }
  __syncthreads();
  for (int idx = tid; idx < 64 * 128; idx += 256) {
    const int m = idx >> 7;
    dots[idx] = expf(dots[idx] - lseS[m]);
  }
  __syncthreads();

  const int nOff2 = (wave & 1) * 32;  // wave owns 2 N-tiles of 64-wide output
  v8f o0 = {}, o1 = {};
  for (int k = 0; k < 128; k += 4) {
    const int kk = k + 2 * half;
    v2f a; a.x = dots[mRow * 128 + kk]; a.y = dots[mRow * 128 + kk + 1];
    const int nc = nOff2 + l16;
    v2f b;
    b.x = vbuf[kk * DH + nc];      b.y = vbuf[(kk + 1) * DH + nc];
    o0 = wmma_f32x4(a, b, o0);
    b.x = vbuf[kk * DH + nc + 16]; b.y = vbuf[(kk + 1) * DH + nc + 16];
    o1 = wmma_f32x4(a, b, o1);
  }
#pragma unroll
  for (int rr = 0; rr < 8; ++rr) {
    const int m = mT * 16 + rr + 8 * half;
    const unsigned tok = stkv[m];
    const float w = wS[m];
    float* dst = ctx + ((size_t)bh * NTOK + tok) * DH + nOff2 + l16;
    dst[0]  += w * o0[rr];   // race-free: one round per launch, disjoint (tok,e)
    dst[16] += w * o1[rr];
  }
}

// ---------------- stage 5: per-token round weights ----------------
__global__ void weight_kernel(const float* __restrict__ logits,
                              float* __restrict__ weight) {
  const int gid = blockIdx.x * blockDim.x + threadIdx.x;  // BH*NTOK
  const int bh = gid >> 12, t = gid & (NTOK - 1);
  float l[NHASH];
  float mx = -3.4e38f;
#pragma unroll
  for (int r = 0; r < NHASH; ++r) {
    l[r] = logits[((size_t)(bh * NHASH + r) * NTOK) + t];
    mx = fmaxf(mx, l[r]);
  }
  float s = 0.0f;
#pragma unroll
  for (int r = 0; r < NHASH; ++r) { l[r] = expf(l[r] - mx); s += l[r]; }
  const float inv = 1.0f / s;
#pragma unroll
  for (int r = 0; r < NHASH; ++r)
    weight[((size_t)(bh * NHASH + r) * NTOK) + t] = l[r] * inv;
}

// ---------------- stage 7: output projection ----------------
__global__ void gemm_out(const float* __restrict__ ctx,
                         const float* __restrict__ Wout,
                         const float* __restrict__ bout,
                         float* __restrict__ out) {
  const int lane = threadIdx.x & 31;
  const int wave = threadIdx.x >> 5;
  const int half = lane >> 4;
  const int l16 = lane & 15;
  const int mBase = blockIdx.x * 16;
  const int nBase = blockIdx.y * 128 + wave * 16;

  const int mA = mBase + l16;
  const int n = nBase + l16;
  const int bidx = mA / NTOK, t = mA % NTOK;

  v8f acc = {};
  for (int k = 0; k < DM; k += 4) {
    const int kk = k + 2 * half;
    const int h = kk >> 6, e = kk & 63;  // un-merge (h, e); e<=62 so e+1 same head
    const float* __restrict__ ap =
        ctx + (((size_t)bidx * NH + h) * NTOK + t) * DH + e;
    v2f a; a.x = ap[0]; a.y = ap[1];
    v2f b;
    b.x = Wout[(size_t)kk * DM + n];
    b.y = Wout[(size_t)(kk + 1) * DM + n];
    acc = wmma_f32x4(a, b, acc);
  }
  const float bias = bout[n];
#pragma unroll
  for (int r = 0; r < 8; ++r) {
    const int m = mBase + r + 8 * half;
    out[(size_t)m * DM + n] = acc[r] + bias;
  }
}

// ---------------- launch ----------------
extern "C" void kernel_launch(void* const* d_in, const int* in_sizes, int n_in,
                              void* d_out, int out_size, void* d_ws,
                              size_t ws_size, hipStream_t stream) {
  const float* queries = (const float*)d_in[0];
  // d_in[1] keys, d_in[2] values: unused by the module
  const float* W_qk = (const float*)d_in[3];
  const float* W_v  = (const float*)d_in[4];
  const float* W_out = (const float*)d_in[5];
  const float* b_out = (const float*)d_in[6];
  const float* rot = (const float*)d_in[7];
  float* out = (float*)d_out;

  char* ws = (char*)d_ws;
  float* qk_m = (float*)(ws + 0);                 // 32*4096*64 f32 = 32MB
  float* v_m  = (float*)(ws + (size_t)33554432);  // 32MB
  float* ctx  = (float*)(ws + (size_t)67108864);  // 32MB
  unsigned* buckets = (unsigned*)(ws + (size_t)100663296);  // 2MB
  unsigned* st      = (unsigned*)(ws + (size_t)102760448);  // 2MB
  float* logits = (float*)(ws + (size_t)104857600);         // 2MB
  float* weight = (float*)(ws + (size_t)106954752);         // 2MB

  // 1) zero accumulation buffer
  zero_f32<<<(BHD * NTOK * DH) / 256, 256, 0, stream>>>(ctx, BHD * NTOK * DH);

  // 2) qk/v projections (f32 WMMA)
  dim3 gq(BATCH * NTOK / 16, DM / 128, 2);
  gemm_qkv<<<gq, 256, 0, stream>>>(queries, W_qk, W_v, qk_m, v_m);

  // 3) LSH hashing
  hash_kernel<<<(BHD * NTOK) / 256, 256, 0, stream>>>(qk_m, rot, buckets);

  // 4) stable counting sort per (bh, round)
  sort_kernel<<<BHD * NHASH, 64, 0, stream>>>(buckets, st);

  // 5) phase A: per-row logsumexp of masked dots (f32 WMMA)
  const size_t smemA = (size_t)12288 * 4 + 128 * 4;
  chunk_kernel<false><<<BHD * NCHUNK, 256, smemA, stream>>>(
      qk_m, v_m, st, nullptr, logits, nullptr, ctx, NCHUNK, 0);

  // 6) cross-round softmax weights
  weight_kernel<<<(BHD * NTOK) / 256, 256, 0, stream>>>(logits, weight);

  // 7) phase B: one launch per round (deterministic, race-free += into ctx)
  const size_t smemB = (size_t)20480 * 4 + (128 + 64 + 64) * 4;
  for (int r = 0; r < NHASH; ++r) {
    chunk_kernel<true><<<BHD * CPR, 256, smemB, stream>>>(
        qk_m, v_m, st, logits, nullptr, weight, ctx, CPR, r * CPR);
  }

  // 8) output projection + bias (f32 WMMA)
  dim3 go(BATCH * NTOK / 16, DM / 128, 1);
  gemm_out<<<go, 256, 0, stream>>>(ctx, W_out, b_out, out);
}